// index2input_17317308137668
// MI455X (gfx1250) — compile-verified
//
#include <hip/hip_runtime.h>
#include <hip/hip_bf16.h>
#include <stdint.h>

// Problem constants (match reference harness exactly).
#define VOCAB      50257   // MAX_WORDS
#define INPUT_SIZE 1024

// Native ext-vector types (the nontemporal builtins require these, not
// HIP_vector_type structs).
typedef float        f32x4 __attribute__((ext_vector_type(4)));
typedef unsigned int u32x4 __attribute__((ext_vector_type(4)));
typedef int          i32x8 __attribute__((ext_vector_type(8)));
typedef int          i32x4 __attribute__((ext_vector_type(4)));

// ---------------------------------------------------------------------------
// Main kernel: embedding gather + bias.
//   out[t, i] = W[i * VOCAB + x[t]] + b[i]
// One token per block; 256 threads; each thread produces 4 consecutive i
// (b128 coalesced write). The 4 W reads share one base address and use
// 24-bit immediate offsets (VOCAB*4, 2*VOCAB*4, 3*VOCAB*4 all < 8MB).
// Output is streamed with NT stores so the 64MB write stream does not evict
// the ~200MB W working set from the 192MB L2.
// ---------------------------------------------------------------------------
__global__ __launch_bounds__(256) void embed_gather_kernel(
    const int* __restrict__ x, const float* __restrict__ W,
    const float* __restrict__ b, float* __restrict__ out) {
  const int tk = blockIdx.x;          // token index
  const int idx = x[tk];              // uniform per block -> scalar load
  const int i0 = threadIdx.x << 2;    // 4 rows of W per thread

  const float* p = W + (size_t)i0 * VOCAB + idx;
  float a0 = p[0];
  float a1 = p[VOCAB];
  float a2 = p[2 * VOCAB];
  float a3 = p[3 * VOCAB];

  f32x4 bb = *(const f32x4*)(b + i0);

  f32x4 r;
  r.x = a0 + bb.x;
  r.y = a1 + bb.y;
  r.z = a2 + bb.z;
  r.w = a3 + bb.w;

  // Non-temporal: output is write-once, keep it out of L2.
  __builtin_nontemporal_store(r, (f32x4*)(out + (size_t)tk * INPUT_SIZE + i0));
}

// ---------------------------------------------------------------------------
// CDNA5 path probe kernel (NOT launched; emitted for compile/ISA feedback).
// Toolchain identified as clang-23/therock lane (6-arg TDM builtin).
// Exercises: tensor_load_to_lds, s_wait_tensorcnt, async global->LDS copy
// via inline asm + s_wait_asynccnt, and the cluster barrier.
// ---------------------------------------------------------------------------
__global__ void cdna5_probe_kernel(const float* __restrict__ g,
                                   float* __restrict__ o) {
  __shared__ float lds[256];

  // TDM descriptor (zero-filled; kernel is never launched).
  u32x4 g0 = {};
  i32x8 g1 = {};
  i32x4 g2 = {};
  i32x4 g3 = {};
  i32x8 g4 = {};
  __builtin_amdgcn_tensor_load_to_lds(g0, g1, g2, g3, g4, 0);
  __builtin_amdgcn_s_wait_tensorcnt(0);

  // Async global->LDS copy of one dword per lane.
  unsigned int ldsoff = (unsigned int)(size_t)&lds[threadIdx.x];
  const float* ga = g + threadIdx.x;
  asm volatile("global_load_async_to_lds_b32 %0, %1, off"
               :: "v"(ldsoff), "v"(ga) : "memory");
  asm volatile("s_wait_asynccnt 0" ::: "memory");

  if (__builtin_amdgcn_cluster_id_x() > 0) {
    __builtin_amdgcn_s_cluster_barrier();
  }
  __syncthreads();
  o[threadIdx.x] = lds[threadIdx.x];
}

// ---------------------------------------------------------------------------
// Launch
// ---------------------------------------------------------------------------
extern "C" void kernel_launch(void* const* d_in, const int* in_sizes, int n_in,
                              void* d_out, int out_size, void* d_ws, size_t ws_size,
                              hipStream_t stream) {
  const int*   x = (const int*)d_in[0];    // [8*2048] int32 token ids
  const float* W = (const float*)d_in[1];  // [1024, 50257] row-major
  const float* b = (const float*)d_in[2];  // [1024]
  float* out = (float*)d_out;              // [8*2048, 1024]

  const int n_tokens = in_sizes[0];        // 16384
  embed_gather_kernel<<<n_tokens, 256, 0, stream>>>(x, W, b, out);
}